// DirGCNConv_weighted_60584808678017
// MI455X (gfx1250) — compile-verified
//
#include <hip/hip_runtime.h>
#include <hip/hip_bf16.h>

// ---------------------------------------------------------------------------
// DirGCNConv (weighted) on gfx1250 (MI455X, CDNA5, wave32)
//   out = 0.5*(norm_spmm_fwd(x) @ Wsrc^T + b_src)
//       + 0.5*(norm_spmm_bwd(x) @ Wdst^T + b_dst)
// Pipeline: zero -> degrees (f32 atomics) -> w_norm -> scatter SpMM (f32
// atomics) -> fused dual-GEMM epilogue using V_WMMA_F32_16X16X4_F32.
// ---------------------------------------------------------------------------

typedef __attribute__((ext_vector_type(2))) float v2f;
typedef __attribute__((ext_vector_type(8))) float v8f;

__device__ __forceinline__ void atomAddF(float* p, float v) {
#if defined(__HIP_DEVICE_COMPILE__)
    unsafeAtomicAdd(p, v);   // hardware global_atomic_add_f32 (device scope)
#else
    (void)p; (void)v;
#endif
}

// ---- kernel 1: zero scratch ------------------------------------------------
__global__ void k_zero(float* __restrict__ p, long long n) {
    long long i = (long long)blockIdx.x * blockDim.x + threadIdx.x;
    if (i < n) p[i] = 0.0f;
}

// ---- kernel 2: weighted degree accumulation --------------------------------
__global__ void k_degree(const int* __restrict__ ei, const float* __restrict__ w,
                         float* __restrict__ outdeg, float* __restrict__ indeg, int E) {
    int e = blockIdx.x * blockDim.x + threadIdx.x;
    if (e >= E) return;
    float we = w[e];
    atomAddF(&outdeg[ei[e]], we);      // row = ei[0, e]
    atomAddF(&indeg[ei[E + e]], we);   // col = ei[1, e]
}

// ---- kernel 3: symmetric normalized edge weight ----------------------------
__global__ void k_wnorm(const int* __restrict__ ei, const float* __restrict__ w,
                        const float* __restrict__ outdeg, const float* __restrict__ indeg,
                        float* __restrict__ wn, int E) {
    int e = blockIdx.x * blockDim.x + threadIdx.x;
    if (e >= E) return;
    float od = outdeg[ei[e]];
    float id = indeg[ei[E + e]];
    float a = (od > 0.0f) ? rsqrtf(od) : 0.0f;
    float b = (id > 0.0f) ? rsqrtf(id) : 0.0f;
    wn[e] = w[e] * a * b;
}

// ---- kernel 4: scatter SpMM (both directions) ------------------------------
// 32 lanes per edge, 2 features per lane (float2 loads, 2 f32 atomics each way)
__global__ void k_scatter(const float* __restrict__ x, const int* __restrict__ ei,
                          const float* __restrict__ wn,
                          float* __restrict__ aggf, float* __restrict__ aggb, int E) {
    long long t = (long long)blockIdx.x * blockDim.x + threadIdx.x;
    int e = (int)(t >> 5);
    int d = (int)(t & 31) * 2;            // feature pair
    if (e >= E) return;
    int row = ei[e];
    int col = ei[E + e];
    float w = wn[e];
    const v2f xc = *(const v2f*)(x + (long long)col * 64 + d);
    const v2f xr = *(const v2f*)(x + (long long)row * 64 + d);
    float* pf = aggf + (long long)row * 64 + d;
    float* pb = aggb + (long long)col * 64 + d;
    atomAddF(pf + 0, w * xc.x);
    atomAddF(pf + 1, w * xc.y);
    atomAddF(pb + 0, w * xr.x);
    atomAddF(pb + 1, w * xr.y);
}

// ---- kernel 5: fused dual GEMM epilogue with WMMA --------------------------
// One wave32 computes one 16x16 tile of out.
//   acc = aggf_tile(16xK) x WsrcT_tile(Kx16) + aggb_tile x WdstT_tile, K=64
//   out = 0.5*acc + 0.5*(b_src + b_dst)
// VGPR layouts per CDNA5 ISA 7.12.2 (16x16x4 f32):
//   A: lane l, vgpr v -> A[M = l%16][K = 4*kk + 2*(l/16) + v]   (float2/lane)
//   B: lane l, vgpr v -> B[K = 4*kk + 2*(l/16) + v][N = l%16]
//      with B[k][n] = W[n][k] (W row-major [64][64]) -> contiguous float2
//   C/D: lane l, vgpr r -> D[M = r + 8*(l/16)][N = l%16]
__global__ void __launch_bounds__(256)
k_wmma_epilogue(const float* __restrict__ aggf, const float* __restrict__ aggb,
                const float* __restrict__ Wsrc, const float* __restrict__ Wdst,
                const float* __restrict__ bsrc, const float* __restrict__ bdst,
                float* __restrict__ out, int tilesM, int Nnodes) {
    int wave  = (int)((blockIdx.x * blockDim.x + threadIdx.x) >> 5);
    int lane  = threadIdx.x & 31;
    int tileM = wave >> 2;         // D_OUT = 64 -> 4 N-tiles
    int tileN = wave & 3;
    if (tileM >= tilesM) return;

    int lmod = lane & 15;
    int lhi  = lane >> 4;          // 0 or 1 -> K offset 0 or 2

    int arow = tileM * 16 + lmod;
    if (arow >= Nnodes) arow = Nnodes - 1;               // clamp (no partials here)
    const float* af_p = aggf + (long long)arow * 64 + 2 * lhi;
    const float* ab_p = aggb + (long long)arow * 64 + 2 * lhi;
    const float* bs_p = Wsrc + (tileN * 16 + lmod) * 64 + 2 * lhi;
    const float* bd_p = Wdst + (tileN * 16 + lmod) * 64 + 2 * lhi;

    v8f acc = {};
#pragma unroll
    for (int kk = 0; kk < 16; ++kk) {
        v2f af = *(const v2f*)(af_p + 4 * kk);
        v2f bs = *(const v2f*)(bs_p + 4 * kk);
        acc = __builtin_amdgcn_wmma_f32_16x16x4_f32(
                  false, af, false, bs, (short)0, acc, false, false);
        v2f ab = *(const v2f*)(ab_p + 4 * kk);
        v2f bd = *(const v2f*)(bd_p + 4 * kk);
        acc = __builtin_amdgcn_wmma_f32_16x16x4_f32(
                  false, ab, false, bd, (short)0, acc, false, false);
    }

    int ncol = tileN * 16 + lmod;
    float bias = 0.5f * (bsrc[ncol] + bdst[ncol]);
    int mbase = tileM * 16 + 8 * lhi;
    float* ocol = out + ncol;
#pragma unroll
    for (int r = 0; r < 8; ++r) {
        int m = mbase + r;
        if (m < Nnodes) ocol[(long long)m * 64] = 0.5f * acc[r] + bias;
    }
}

// ---------------------------------------------------------------------------
extern "C" void kernel_launch(void* const* d_in, const int* in_sizes, int n_in,
                              void* d_out, int out_size, void* d_ws, size_t ws_size,
                              hipStream_t stream) {
    const float* x    = (const float*)d_in[0];   // [N, 64]
    const int*   ei   = (const int*)d_in[1];     // [2, E]
    const float* ew   = (const float*)d_in[2];   // [E]
    const float* Wsrc = (const float*)d_in[3];   // [64, 64]
    const float* bsrc = (const float*)d_in[4];   // [64]
    const float* Wdst = (const float*)d_in[5];   // [64, 64]
    const float* bdst = (const float*)d_in[6];   // [64]
    float* out = (float*)d_out;

    const int N = in_sizes[0] / 64;
    const int E = in_sizes[2];

    // Workspace layout (f32):
    //   [0,N)          outdeg
    //   [N,2N)         indeg
    //   [2N,2N+E)      w_norm
    //   [..,+64N)      agg_fwd
    //   [..,+64N)      agg_bwd
    float* ws     = (float*)d_ws;
    float* outdeg = ws;
    float* indeg  = ws + (long long)N;
    float* wn     = ws + 2LL * N;
    float* aggf   = wn + (long long)E;
    float* aggb   = aggf + 64LL * N;

    const long long ztotal = 130LL * N + (long long)E;
    k_zero<<<(unsigned)((ztotal + 255) / 256), 256, 0, stream>>>(ws, ztotal);

    const int eb = (E + 255) / 256;
    k_degree<<<eb, 256, 0, stream>>>(ei, ew, outdeg, indeg, E);
    k_wnorm<<<eb, 256, 0, stream>>>(ei, ew, outdeg, indeg, wn, E);

    const long long st = 32LL * E;
    k_scatter<<<(unsigned)((st + 255) / 256), 256, 0, stream>>>(x, ei, wn, aggf, aggb, E);

    const int tilesM = (N + 15) / 16;
    const int waves  = tilesM * 4;              // 4 tiles along D_OUT=64
    k_wmma_epilogue<<<(waves + 7) / 8, 256, 0, stream>>>(aggf, aggb, Wsrc, Wdst,
                                                         bsrc, bdst, out, tilesM, N);
}